// lstm_simp_10187662426387
// MI455X (gfx1250) — compile-verified
//
#include <hip/hip_runtime.h>

typedef __attribute__((ext_vector_type(2))) float v2f;
typedef __attribute__((ext_vector_type(8))) float v8f;

#define T_STEPS 250
#define BATCH   250

__device__ __forceinline__ float sigmoidf_(float x) {
    return 1.0f / (1.0f + __expf(-x));
}
// tanh(x) = 1 - 2/(exp(2x)+1); saturates correctly to +/-1 for large |x|
__device__ __forceinline__ float tanhf_(float x) {
    return 1.0f - 2.0f / (__expf(2.0f * x) + 1.0f);
}

// One encoder LSTM step for this lane's batch element. PyTorch gate order i,f,g,o.
__device__ __forceinline__ void enc_step(float xv,
                                         const float wih[16], const float bv[16],
                                         const float whh[16][4],
                                         float h[4], float c[4]) {
    float gt[16];
#pragma unroll
    for (int g = 0; g < 16; ++g) {
        float s = xv * wih[g] + bv[g];
#pragma unroll
        for (int k = 0; k < 4; ++k) s += h[k] * whh[g][k];
        gt[g] = s;
    }
#pragma unroll
    for (int k = 0; k < 4; ++k) {
        float cn = sigmoidf_(gt[4 + k]) * c[k] + sigmoidf_(gt[k]) * tanhf_(gt[8 + k]);
        c[k] = cn;
        h[k] = sigmoidf_(gt[12 + k]) * tanhf_(cn);
    }
}

__global__ __launch_bounds__(256, 1)
void lstm_fused_kernel(const float* __restrict__ x,        // (250,250) -> (T,B,1)
                       const float* __restrict__ h0_dec,   // (250,1)
                       const float* __restrict__ enc_w_ih, // (16,1)
                       const float* __restrict__ enc_w_hh, // (16,4)
                       const float* __restrict__ enc_b,    // (16,)
                       const float* __restrict__ fc1_w,    // (512,1000)
                       const float* __restrict__ fc1_b,    // (512,)
                       const float* __restrict__ fc2_w,    // (250,512)
                       const float* __restrict__ fc2_b,    // (250,)
                       const float* __restrict__ dec_w_ih, // (4,4)
                       const float* __restrict__ dec_w_hh, // (4,1)
                       const float* __restrict__ dec_b,    // (4,)
                       float* __restrict__ out)            // (250,250,1)
{
    __shared__ __align__(16) float lds_cell[1000]; // encoder final cell, (b*4+k)
    __shared__ __align__(16) float lds_fc1[512];   // relu(fc1) activations

    const int  tid    = threadIdx.x;
    const bool active = tid < BATCH;

    // Warm L2 with fc1_w while the serial encoder scan runs (global_prefetch_b8).
    __builtin_prefetch(fc1_w + (size_t)tid * 1000, 0, 1);

    // ---- encoder weights into VGPRs (uniform -> SMEM loads) ----
    float wih[16], bv[16], whh[16][4];
#pragma unroll
    for (int g = 0; g < 16; ++g) {
        wih[g] = enc_w_ih[g];
        bv[g]  = enc_b[g];
#pragma unroll
        for (int k = 0; k < 4; ++k) whh[g][k] = enc_w_hh[g * 4 + k];
    }

    // ================= Pass A: encoder scan, keep only final cell =================
    float h[4] = {0.f, 0.f, 0.f, 0.f};
    float c[4] = {0.f, 0.f, 0.f, 0.f};
    if (active) {
        for (int t = 0; t < T_STEPS; ++t)
            enc_step(x[t * BATCH + tid], wih, bv, whh, h, c);
#pragma unroll
        for (int k = 0; k < 4; ++k) lds_cell[tid * 4 + k] = c[k];
    }
    __syncthreads();

    // ================= Phase B: fc1 via V_WMMA_F32_16X16X4_F32 =================
    // out1 = relu(stacked(1x1000) @ fc1_w.T + fc1_b). M=16 tile, only row M=0 live.
    // A 16x4 f32 layout: lanes0-15 v0/v1 = A[M=lane, K=0/1]; lanes16-31 = K=2/3.
    // B 4x16 layout (mirrors A/C pattern): reg = K&1, lane-half = K>>1, N = lane&15.
    // C 16x16: VGPR r, lanes0-15 -> M=r, N=lane  => row 0 = acc[0] on lanes 0..15.
    const int  wave   = tid >> 5;
    const int  lane   = tid & 31;
    const int  ncol   = lane & 15;
    const int  khalf  = (lane >> 4) << 1;  // 0 or 2
    const bool a_keep = (ncol == 0);       // only lanes 0 and 16 carry A row M=0
#pragma unroll 1
    for (int rep = 0; rep < 4; ++rep) {
        const int n0  = (wave * 4 + rep) * 16;
        const int nmy = n0 + ncol;
        const float* wrow = fc1_w + (size_t)nmy * 1000 + khalf;
        v8f acc = {0.f, 0.f, 0.f, 0.f, 0.f, 0.f, 0.f, 0.f};
        for (int kk = 0; kk < 1000; kk += 4) {
            // EXEC-uniform A fetch: lanes 0-15 broadcast {s[kk],s[kk+1]},
            // lanes 16-31 broadcast {s[kk+2],s[kk+3]}; 8B-aligned ds_load_b64.
            const v2f raw = *(const v2f*)&lds_cell[kk + khalf];
            v2f a;
            a.x = a_keep ? raw.x : 0.0f;   // loop-invariant VCC -> v_cndmask
            a.y = a_keep ? raw.y : 0.0f;
            const v2f bm = *(const v2f*)(wrow + kk);  // 8B-aligned global_load_b64
            acc = __builtin_amdgcn_wmma_f32_16x16x4_f32(
                false, a, false, bm, (short)0, acc, false, false);
        }
        if (lane < 16) {
            float v = acc[0] + fc1_b[nmy];
            lds_fc1[nmy] = v > 0.f ? v : 0.f;
        }
    }
    __syncthreads();

    // ================= fc2: per-lane GEMV -> decoder initial cell =================
    float c_dec = 0.f, h_dec = 0.f;
    if (active) {
        float s = fc2_b[tid];
        const float* w2 = fc2_w + (size_t)tid * 512;
        for (int j = 0; j < 512; ++j) s += lds_fc1[j] * w2[j];
        c_dec = s;
        h_dec = h0_dec[tid];
    }

    // ======== Pass C: re-run encoder fused with decoder (y never hits memory) =====
    if (active) {
        float dwih[16], dwhh[4], db[4];
#pragma unroll
        for (int g = 0; g < 4; ++g) {
            db[g]   = dec_b[g];
            dwhh[g] = dec_w_hh[g];
#pragma unroll
            for (int j = 0; j < 4; ++j) dwih[g * 4 + j] = dec_w_ih[g * 4 + j];
        }
#pragma unroll
        for (int k = 0; k < 4; ++k) { h[k] = 0.f; c[k] = 0.f; }

        for (int t = 0; t < T_STEPS; ++t) {
            // encoder step -> h is y[t, b, :]
            enc_step(x[t * BATCH + tid], wih, bv, whh, h, c);
            // decoder step (hidden = 1), gate order i,f,g,o
            float g0 = db[0], g1 = db[1], g2 = db[2], g3 = db[3];
#pragma unroll
            for (int j = 0; j < 4; ++j) {
                g0 += h[j] * dwih[0 * 4 + j];
                g1 += h[j] * dwih[1 * 4 + j];
                g2 += h[j] * dwih[2 * 4 + j];
                g3 += h[j] * dwih[3 * 4 + j];
            }
            g0 += h_dec * dwhh[0];
            g1 += h_dec * dwhh[1];
            g2 += h_dec * dwhh[2];
            g3 += h_dec * dwhh[3];
            c_dec = sigmoidf_(g1) * c_dec + sigmoidf_(g0) * tanhf_(g2);
            h_dec = sigmoidf_(g3) * tanhf_(c_dec);
            out[t * BATCH + tid] = h_dec;
        }
    }
}

extern "C" void kernel_launch(void* const* d_in, const int* in_sizes, int n_in,
                              void* d_out, int out_size, void* d_ws, size_t ws_size,
                              hipStream_t stream) {
    (void)in_sizes; (void)n_in; (void)out_size; (void)d_ws; (void)ws_size;
    lstm_fused_kernel<<<1, 256, 0, stream>>>(
        (const float*)d_in[0],  // x
        (const float*)d_in[1],  // h0_dec
        (const float*)d_in[2],  // enc_w_ih
        (const float*)d_in[3],  // enc_w_hh
        (const float*)d_in[4],  // enc_b
        (const float*)d_in[5],  // fc1_w
        (const float*)d_in[6],  // fc1_b
        (const float*)d_in[7],  // fc2_w
        (const float*)d_in[8],  // fc2_b
        (const float*)d_in[9],  // dec_w_ih
        (const float*)d_in[10], // dec_w_hh
        (const float*)d_in[11], // dec_b
        (float*)d_out);
}